// GCNConv_58128087384147
// MI455X (gfx1250) — compile-verified
//
#include <hip/hip_runtime.h>
#include <hip/hip_bf16.h>

typedef __attribute__((ext_vector_type(2))) float v2f;
typedef __attribute__((ext_vector_type(8))) float v8f;

#define D_ROWS 128
#define F_IN   1024
#define N_OUT  32768
#define KC     128
#define LDSW   130   // padded row stride (floats) to break 64-bank conflicts

// ---------------------------------------------------------------------------
// GEMM:  yT[n, d] = sum_k W[n,k] * x[d,k]     (yT is [N, 128])
// grid.x = N/16 blocks, 256 threads = 8 waves; wave w owns d-tile [16w,16w+16)
// A (SRC0) = W tile 16(M=n) x 4(K)  from LDS
// B (SRC1) = xT tile 4(K) x 16(N=d) from global (x is L2-resident, 512 KB)
// ---------------------------------------------------------------------------
__global__ __launch_bounds__(256) void gcn_gemm_wmma(
    const float* __restrict__ x, const float* __restrict__ W,
    float* __restrict__ yT) {
  __shared__ float wtile[16 * LDSW];

  const int n0    = blockIdx.x << 4;        // 16 W-rows per block
  const int wave  = threadIdx.x >> 5;       // 0..7  -> d tile
  const int lane  = threadIdx.x & 31;
  const int d0    = wave << 4;
  const int lhalf = lane >> 4;              // 0 or 1 (K split across halves)
  const int lmod  = lane & 15;

  // staging: thread t loads W row (t/16), cols (t%16)*8 .. +7  (2x b128)
  const int srow = threadIdx.x >> 4;        // 0..15
  const int scol = (threadIdx.x & 15) << 3; // 0,8,...,120

  v8f acc = {};

  for (int k0 = 0; k0 < F_IN; k0 += KC) {
    const float* gsrc = W + (size_t)(n0 + srow) * F_IN + (k0 + scol);
    float4 wa = *(const float4*)(gsrc);
    float4 wb = *(const float4*)(gsrc + 4);
    if (k0 + KC < F_IN) {
      __builtin_prefetch(gsrc + KC, 0, 1);   // global_prefetch next chunk
    }
    float* dst = &wtile[srow * LDSW + scol];
    *(float4*)(dst)     = wa;
    *(float4*)(dst + 4) = wb;
    __syncthreads();

#pragma unroll 8
    for (int kk = 0; kk < KC; kk += 4) {
      // A frag (16x4 f32): lane lmod -> row n0+lmod; VGPR0/1 = K {kk,kk+1} for
      // lanes 0-15, K {kk+2,kk+3} for lanes 16-31.
      const float* ap = &wtile[lmod * LDSW + kk + (lhalf << 1)];
      v2f afrag; afrag.x = ap[0]; afrag.y = ap[1];
      // B frag (4x16 f32): lane lmod -> col d0+lmod, mirrored K split.
      const float* bp = x + (size_t)(d0 + lmod) * F_IN + k0 + kk + (lhalf << 1);
      v2f bfrag; bfrag.x = bp[0]; bfrag.y = bp[1];
      acc = __builtin_amdgcn_wmma_f32_16x16x4_f32(
          /*neg_a=*/false, afrag, /*neg_b=*/false, bfrag,
          /*c_mod=*/(short)0, acc, /*reuse_a=*/false, /*reuse_b=*/false);
    }
    __syncthreads();
  }

  // D layout: VGPR g, lane -> row n = n0 + g + 8*lhalf, col d = d0 + lmod.
  // Store coalesced into yT[n*128 + d] (lanes 0-15 contiguous 64B).
#pragma unroll
  for (int g = 0; g < 8; ++g) {
    yT[(size_t)(n0 + g + (lhalf << 3)) * D_ROWS + d0 + lmod] = acc[g];
  }
}

// ---------------------------------------------------------------------------
// Zero-fill helper (acc must be zeroed every call; harness doesn't re-poison)
// ---------------------------------------------------------------------------
__global__ __launch_bounds__(256) void gcn_zero(float* __restrict__ p, int n4) {
  int i = blockIdx.x * blockDim.x + threadIdx.x;
  if (i < n4) ((float4*)p)[i] = make_float4(0.f, 0.f, 0.f, 0.f);
}

// ---------------------------------------------------------------------------
// Scatter: acc[c, :] += v * yT[r, :]   (one wave per nonzero; lane handles 4
// contiguous d -> 512B coalesced load + line-local f32 atomics)
// ---------------------------------------------------------------------------
__global__ __launch_bounds__(256) void gcn_scatter_nT(
    const float* __restrict__ yT, const int* __restrict__ rows,
    const int* __restrict__ cols, const float* __restrict__ vals,
    float* __restrict__ acc, int nnz) {
  int e    = (blockIdx.x * blockDim.x + threadIdx.x) >> 5;
  int lane = threadIdx.x & 31;
  if (e >= nnz) return;
  int   r = rows[e];
  int   c = cols[e];
  float v = vals[e];
  const float4 yv = *(const float4*)(yT + (size_t)r * D_ROWS + (lane << 2));
  float* op = acc + (size_t)c * D_ROWS + (lane << 2);
  atomicAdd(op + 0, yv.x * v);
  atomicAdd(op + 1, yv.y * v);
  atomicAdd(op + 2, yv.z * v);
  atomicAdd(op + 3, yv.w * v);
}

// Fallback scatter straight into out[d*N + c] ([D,N] layout, slower atomics)
__global__ __launch_bounds__(256) void gcn_scatter_dn(
    const float* __restrict__ yT, const int* __restrict__ rows,
    const int* __restrict__ cols, const float* __restrict__ vals,
    float* __restrict__ out, int nnz) {
  int e    = (blockIdx.x * blockDim.x + threadIdx.x) >> 5;
  int lane = threadIdx.x & 31;
  if (e >= nnz) return;
  int   r = rows[e];
  int   c = cols[e];
  float v = vals[e];
#pragma unroll
  for (int j = 0; j < 4; ++j) {
    int d = lane + (j << 5);
    atomicAdd(out + (size_t)d * N_OUT + c, yT[(size_t)r * D_ROWS + d] * v);
  }
}

// ---------------------------------------------------------------------------
// Tiled transpose: out[d*N + n] = acc[n*128 + d], 32x32 tiles via LDS
// grid = (N/32, D/32)
// ---------------------------------------------------------------------------
__global__ __launch_bounds__(256) void gcn_transpose(
    const float* __restrict__ acc, float* __restrict__ out) {
  __shared__ float tile[32][33];
  int nb = blockIdx.x << 5;
  int db = blockIdx.y << 5;
  int tx = threadIdx.x & 31;
  int ty = threadIdx.x >> 5;   // 0..7
#pragma unroll
  for (int k = 0; k < 4; ++k) {
    int n = nb + ty + (k << 3);
    tile[ty + (k << 3)][tx] = acc[(size_t)n * D_ROWS + db + tx];
  }
  __syncthreads();
#pragma unroll
  for (int k = 0; k < 4; ++k) {
    int d = db + ty + (k << 3);
    out[(size_t)d * N_OUT + nb + tx] = tile[tx][ty + (k << 3)];
  }
}

extern "C" void kernel_launch(void* const* d_in, const int* in_sizes, int n_in,
                              void* d_out, int out_size, void* d_ws, size_t ws_size,
                              hipStream_t stream) {
  const float* x    = (const float*)d_in[0];
  const float* W    = (const float*)d_in[1];
  const int*   rows = (const int*)d_in[2];
  const int*   cols = (const int*)d_in[3];
  const float* vals = (const float*)d_in[4];
  float*       out  = (float*)d_out;
  const int    nnz  = in_sizes[2];

  const size_t ybytes = (size_t)N_OUT * D_ROWS * sizeof(float);  // 16 MB
  float* yT = (float*)d_ws;

  // Stage 1: GEMM yT = W * x^T   (WMMA f32 16x16x4)
  gcn_gemm_wmma<<<N_OUT / 16, 256, 0, stream>>>(x, W, yT);

  const int scatter_blocks = (nnz + 7) / 8;  // 8 waves (nonzeros) per block
  const int n_elem4 = (N_OUT * D_ROWS) / 4;

  if (ws_size >= 2 * ybytes) {
    // Fast path: accumulate transposed [N,128], then tiled transpose.
    float* acc = (float*)((char*)d_ws + ybytes);
    gcn_zero<<<(n_elem4 + 255) / 256, 256, 0, stream>>>(acc, n_elem4);
    gcn_scatter_nT<<<scatter_blocks, 256, 0, stream>>>(yT, rows, cols, vals,
                                                       acc, nnz);
    gcn_transpose<<<dim3(N_OUT / 32, D_ROWS / 32), 256, 0, stream>>>(acc, out);
  } else {
    // Fallback: atomics straight into [D, N] output.
    gcn_zero<<<(n_elem4 + 255) / 256, 256, 0, stream>>>(out, n_elem4);
    gcn_scatter_dn<<<scatter_blocks, 256, 0, stream>>>(yT, rows, cols, vals,
                                                       out, nnz);
  }
}